// NNFOwithBayesianJumps_59889023976081
// MI455X (gfx1250) — compile-verified
//
#include <hip/hip_runtime.h>
#include <cmath>

typedef __attribute__((ext_vector_type(2))) float v2f;
typedef __attribute__((ext_vector_type(8))) float v8f;

#define Bsz   512
#define Tsz   64
#define Dsz   32
#define Hsz   512
#define Psz   32
#define LOG2PI_F 1.8378770664093453f
#define LOG_OBS_F (-4.605170185988091f)   // ln(0.01)

__device__ __forceinline__ v8f v8zero() {
    v8f z = {0.f,0.f,0.f,0.f,0.f,0.f,0.f,0.f};
    return z;
}

__device__ __forceinline__ v8f wmma4(v2f a, v2f b, v8f c) {
    // D = A(16x4,f32) * B(4x16,f32) + C(16x16,f32)
    return __builtin_amdgcn_wmma_f32_16x16x4_f32(false, a, false, b, (short)0, c, false, false);
}

__device__ __forceinline__ float sigmoidf(float x) {
    return 1.0f / (1.0f + __expf(-x));
}

// Wave computes one 16x16 f32 tile.
// A rows from LDS (stride lda, padded so stride % 64 == 4 -> conflict-free frags).
// B from global in K-pair-packed layout: Bp[k2*ldb + n] = {B[2k2][n], B[2k2+1][n]},
// so each lane's B fragment is ONE aligned b64 load.
// ncol already includes (lane & 15).
__device__ __forceinline__ v8f gemm_tile(const float* __restrict__ sA, int lda,
                                         const v2f* __restrict__ Bp, int ldb,
                                         int ncol, int K, int lane, v8f acc) {
    const int mrow = lane & 15;
    const int kh   = (lane >> 4) << 1;     // 0 or 2
    const float* ap = sA + mrow * lda + kh;
    const v2f*   bp = Bp + (size_t)(kh >> 1) * ldb + ncol;
    #pragma unroll 4
    for (int k = 0; k < K; k += 4) {
        v2f a; a.x = ap[0]; a.y = ap[1];
        acc = wmma4(a, bp[0], acc);
        ap += 4;
        bp += 2 * (size_t)ldb;
    }
    return acc;
}

// ---------------------------------------------------------------- utilities
__global__ void k_zero(float* __restrict__ lastt) {
    int i = threadIdx.x;
    if (i < Bsz) lastt[i] = 0.f;
}

// Pack W (N rows x K cols, row-major), used as B = W^T:
//   dst[k2*N + n] = {W[n][2k2], W[n][2k2+1]}
__global__ void k_packT(v2f* __restrict__ dst, const float* __restrict__ src,
                        int N, int K) {
    int idx = blockIdx.x * blockDim.x + threadIdx.x;
    if (idx < N * (K >> 1)) {
        int n  = idx % N;
        int k2 = idx / N;
        v2f v;
        v.x = src[n * K + 2 * k2];
        v.y = src[n * K + 2 * k2 + 1];
        dst[idx] = v;
    }
}

// Pack B (K rows x N cols, row-major), used directly:
//   dst[k2*N + n] = {B[2k2][n], B[2k2+1][n]}
__global__ void k_packB(v2f* __restrict__ dst, const float* __restrict__ src,
                        int K, int N) {
    int idx = blockIdx.x * blockDim.x + threadIdx.x;
    if (idx < N * (K >> 1)) {
        int n  = idx % N;
        int k2 = idx / N;
        v2f v;
        v.x = src[(2 * k2) * N + n];
        v.y = src[(2 * k2 + 1) * N + n];
        dst[idx] = v;
    }
}

// ---------------------------------------------------------------- h0 kernel
// h0 = tanh(relu(cov @ cov_w1 + b1) @ cov_w2 + b2)
__global__ __launch_bounds__(512) void k_h0(const float* __restrict__ cov,
                                            const v2f* __restrict__ w1p,
                                            const float* __restrict__ b1,
                                            const v2f* __restrict__ w2p,
                                            const float* __restrict__ b2,
                                            float* __restrict__ ws_h) {
    __shared__ float sC[16 * 68];
    __shared__ float sC1[16 * 132];
    const int tid   = threadIdx.x;
    const int lane  = tid & 31;
    const int wv    = tid >> 5;
    const int l15   = lane & 15;
    const int mb    = (lane >> 4) << 3;
    const int bbase = blockIdx.x * 16;

    for (int e = tid; e < 16 * 64; e += 512) {
        int r = e >> 6, c = e & 63;
        sC[r * 68 + c] = cov[(bbase + r) * 64 + c];
    }
    __syncthreads();

    if (wv < 8) {                                     // 16x64x128 GEMM
        int ncol = wv * 16 + l15;
        v8f acc = v8zero();
        acc = gemm_tile(sC, 68, w1p, 128, ncol, 64, lane, acc);
        float bias = b1[ncol];
        #pragma unroll
        for (int v = 0; v < 8; ++v)
            sC1[(mb + v) * 132 + ncol] = fmaxf(acc[v] + bias, 0.f);
    }
    __syncthreads();

    for (int tt = 0; tt < 2; ++tt) {                  // 16x128x512 GEMM
        int ncol = (wv + tt * 16) * 16 + l15;
        v8f acc = v8zero();
        acc = gemm_tile(sC1, 132, w2p, 512, ncol, 128, lane, acc);
        float bias = b2[ncol];
        #pragma unroll
        for (int v = 0; v < 8; ++v)
            ws_h[(size_t)(bbase + mb + v) * 512 + ncol] = tanhf(acc[v] + bias);
    }
}

// ------------------------------------------------------------ p_model helper
__device__ __forceinline__ void p_model_dev(const float* __restrict__ sIn,
                                            const v2f* __restrict__ pw1p,
                                            const float* __restrict__ p_b1,
                                            const v2f* __restrict__ pw2p,
                                            const float* __restrict__ p_b2,
                                            float* __restrict__ sP1,
                                            float* __restrict__ sP2,
                                            int wv, int lane) {
    const int l15 = lane & 15;
    const int mb  = (lane >> 4) << 3;
    if (wv < 8) {                                     // 16x512x128, relu
        int ncol = wv * 16 + l15;
        v8f acc = v8zero();
        acc = gemm_tile(sIn, 516, pw1p, 128, ncol, 512, lane, acc);
        float bias = p_b1[ncol];
        #pragma unroll
        for (int v = 0; v < 8; ++v)
            sP1[(mb + v) * 132 + ncol] = fmaxf(acc[v] + bias, 0.f);
    }
    __syncthreads();
    if (wv < 4) {                                     // 16x128x64
        int ncol = wv * 16 + l15;
        v8f acc = v8zero();
        acc = gemm_tile(sP1, 132, pw2p, 64, ncol, 128, lane, acc);
        float bias = p_b2[ncol];
        #pragma unroll
        for (int v = 0; v < 8; ++v)
            sP2[(mb + v) * 68 + ncol] = acc[v] + bias;
    }
    __syncthreads();
}

// ---------------------------------------------------------------- step kernel
__global__ __launch_bounds__(512) void k_step(
    int t,
    const float* __restrict__ times, const float* __restrict__ Xg,
    const float* __restrict__ Mg,    const int*   __restrict__ num_obs,
    const v2f*   __restrict__ pw1p,  const float* __restrict__ p_b1,
    const v2f*   __restrict__ pw2p,  const float* __restrict__ p_b2,
    const float* __restrict__ wprep, const float* __restrict__ bprep,
    const float* __restrict__ gbih,  const float* __restrict__ gbhh,
    const v2f*   __restrict__ whrP,  const v2f*   __restrict__ whzP,
    const v2f*   __restrict__ whhP,  const v2f*   __restrict__ whhGP,
    const v2f*   __restrict__ wihP,
    float* __restrict__ ws_h, float* __restrict__ ws_lastt,
    float* __restrict__ loss_part, float* __restrict__ kl_part) {

    __shared__ float sH [16 * 516];   // h (input)
    __shared__ float sA [16 * 516];   // rh -> h_ode
    __shared__ float sBt[16 * 516];   // u  -> h_new
    __shared__ float sXP[16 * 1028];  // xp (K = 1024)
    __shared__ float sP1[16 * 132];
    __shared__ float sP2[16 * 68];
    __shared__ float sDiff[16];
    __shared__ float sM[16];
    __shared__ float sRed[1];

    const int tid   = threadIdx.x;
    const int lane  = tid & 31;
    const int wv    = tid >> 5;
    const int l15   = lane & 15;
    const int mb    = (lane >> 4) << 3;
    const int bbase = blockIdx.x * 16;

    // load h tile, diff, masks
    for (int e = tid; e < 16 * 512; e += 512) {
        int r = e >> 9, c = e & 511;
        sH[r * 516 + c] = ws_h[(size_t)(bbase + r) * 512 + c];
    }
    if (tid < 16) {
        int bg = bbase + tid;
        float mv = (num_obs[bg] > t) ? 1.f : 0.f;
        float tc = times[bg * Tsz + t];
        sM[tid]    = mv;
        sDiff[tid] = (tc - ws_lastt[bg]) * mv;
    }
    if (tid == 0) sRed[0] = 0.f;
    __syncthreads();

    // Phase A: rh = sigmoid(h @ whr^T) * h           (16x512x512)
    for (int tt = 0; tt < 2; ++tt) {
        int ncol = (wv + tt * 16) * 16 + l15;
        v8f acc = v8zero();
        acc = gemm_tile(sH, 516, whrP, 512, ncol, 512, lane, acc);
        #pragma unroll
        for (int v = 0; v < 8; ++v) {
            float hv = sH[(mb + v) * 516 + ncol];
            sA[(mb + v) * 516 + ncol] = hv * sigmoidf(acc[v]);
        }
    }
    __syncthreads();

    // Phase B: u = tanh(rh @ whh^T)                  (16x512x512)
    for (int tt = 0; tt < 2; ++tt) {
        int ncol = (wv + tt * 16) * 16 + l15;
        v8f acc = v8zero();
        acc = gemm_tile(sA, 516, whhP, 512, ncol, 512, lane, acc);
        #pragma unroll
        for (int v = 0; v < 8; ++v)
            sBt[(mb + v) * 516 + ncol] = tanhf(acc[v]);
    }
    __syncthreads();

    // Phase C: z = sigmoid(h @ whz^T);  h_ode = h + (1-z)*(u-h)*diff  -> sA
    for (int tt = 0; tt < 2; ++tt) {
        int ncol = (wv + tt * 16) * 16 + l15;
        v8f acc = v8zero();
        acc = gemm_tile(sH, 516, whzP, 512, ncol, 512, lane, acc);
        #pragma unroll
        for (int v = 0; v < 8; ++v) {
            float z  = sigmoidf(acc[v]);
            float hv = sH [(mb + v) * 516 + ncol];
            float uv = sBt[(mb + v) * 516 + ncol];
            sA[(mb + v) * 516 + ncol] = hv + (1.f - z) * (uv - hv) * sDiff[mb + v];
        }
    }
    __syncthreads();

    // Phase D: p = p_model(h_ode)
    p_model_dev(sA, pw1p, p_b1, pw2p, p_b2, sP1, sP2, wv, lane);

    // Phase E: losses + feature prep (xp)
    const int eb = tid >> 5, ed = tid & 31;
    const int bg = bbase + eb;
    const size_t xoff = (size_t)bg * (Tsz * Dsz) + (size_t)t * Dsz + ed;
    const float Xc = Xg[xoff];
    const float Mc = Mg[xoff];
    const float mvE = sM[eb];
    {
        float mean = sP2[eb * 68 + ed];
        float logv = sP2[eb * 68 + 32 + ed];
        float sig  = __expf(0.5f * logv);
        float err  = (Xc - mean) / sig;
        float le   = 0.5f * (err * err + logv + LOG2PI_F) * Mc * mvE;
        atomicAdd(&sRed[0], le);
        const float* wp = wprep + ed * 128;
        const float* bp = bprep + ed * 32;
        #pragma unroll 4
        for (int p = 0; p < 32; ++p) {
            float val = wp[p] * Xc + wp[32 + p] * mean + wp[64 + p] * logv
                      + wp[96 + p] * err + bp[p];
            sXP[eb * 1028 + ed * 32 + p] = fmaxf(val, 0.f) * Mc;
        }
    }
    __syncthreads();
    if (tid == 0) { loss_part[t * 32 + blockIdx.x] = sRed[0]; sRed[0] = 0.f; }
    __syncthreads();

    // Phase F: GRU update.  gi = xp @ wih^T, gh = h_ode @ whh_gru^T, gates fused.
    for (int tt = 0; tt < 2; ++tt) {
        const int n0   = (wv + tt * 16) * 16;
        const int ncol = n0 + l15;
        v8f aR = v8zero(), aZg = v8zero(), aNI = v8zero(), aNH = v8zero();
        const int kh = (lane >> 4) << 1;
        {   // xp contribution, K = 1024  (3 packed-b64 B loads per A frag)
            const float* ap = sXP + l15 * 1028 + kh;
            const v2f*   bp = wihP + (size_t)(kh >> 1) * 1536 + n0 + l15;
            #pragma unroll 2
            for (int k = 0; k < 1024; k += 4) {
                v2f a; a.x = ap[0]; a.y = ap[1];
                aR  = wmma4(a, bp[0],    aR);
                aZg = wmma4(a, bp[512],  aZg);
                aNI = wmma4(a, bp[1024], aNI);
                ap += 4; bp += 2 * 1536;
            }
        }
        {   // h_ode contribution, K = 512
            const float* ap = sA + l15 * 516 + kh;
            const v2f*   bp = whhGP + (size_t)(kh >> 1) * 1536 + n0 + l15;
            #pragma unroll 2
            for (int k = 0; k < 512; k += 4) {
                v2f a; a.x = ap[0]; a.y = ap[1];
                aR  = wmma4(a, bp[0],    aR);
                aZg = wmma4(a, bp[512],  aZg);
                aNH = wmma4(a, bp[1024], aNH);
                ap += 4; bp += 2 * 1536;
            }
        }
        const float br  = gbih[ncol] + gbhh[ncol];
        const float bz  = gbih[512 + ncol] + gbhh[512 + ncol];
        const float bni = gbih[1024 + ncol];
        const float bnh = gbhh[1024 + ncol];
        #pragma unroll
        for (int v = 0; v < 8; ++v) {
            float rg = sigmoidf(aR[v] + br);
            float zg = sigmoidf(aZg[v] + bz);
            float ng = tanhf(aNI[v] + bni + rg * (aNH[v] + bnh));
            float hp = sA[(mb + v) * 516 + ncol];
            float hn = (1.f - zg) * ng + zg * hp;
            sBt[(mb + v) * 516 + ncol] = (sM[mb + v] > 0.f) ? hn : hp;
        }
    }
    __syncthreads();

    // Phase G: p2 = p_model(h_new), KL
    p_model_dev(sBt, pw1p, p_b1, pw2p, p_b2, sP1, sP2, wv, lane);
    {
        float mean2 = sP2[eb * 68 + ed];
        float logv2 = sP2[eb * 68 + 32 + ed];
        float var2  = __expf(logv2);
        float dm    = mean2 - Xc;
        float kle = (LOG_OBS_F - 0.5f * logv2 + (var2 + dm * dm) * 5000.0f - 0.5f)
                    * Mc * mvE;
        atomicAdd(&sRed[0], kle);
    }
    __syncthreads();
    if (tid == 0) kl_part[t * 32 + blockIdx.x] = sRed[0];

    // write back h, last_t
    for (int e = tid; e < 16 * 512; e += 512) {
        int r = e >> 9, c = e & 511;
        ws_h[(size_t)(bbase + r) * 512 + c] = sBt[r * 516 + c];
    }
    if (tid < 16 && sM[tid] > 0.f)
        ws_lastt[bbase + tid] = times[(bbase + tid) * Tsz + t];
}

// ---------------------------------------------------------------- finalize
__global__ void k_final(const float* __restrict__ ws_h,
                        const float* __restrict__ loss_part,
                        const float* __restrict__ kl_part,
                        const int*   __restrict__ num_obs,
                        float* __restrict__ out) {
    size_t tid = (size_t)blockIdx.x * blockDim.x + threadIdx.x;
    size_t stride = (size_t)gridDim.x * blockDim.x;
    for (size_t i = tid; i < (size_t)Bsz * Hsz; i += stride) out[i] = ws_h[i];
    if (tid == 0) {
        float l1 = 0.f, l2 = 0.f, nll = 0.f;
        for (int t = 0; t < Tsz; ++t) {
            float lp = 0.f, kp = 0.f;
            for (int b = 0; b < 32; ++b) {
                lp += loss_part[t * 32 + b];
                kp += kl_part[t * 32 + b];
            }
            int sm = 0;
            for (int b = 0; b < Bsz; ++b) sm += (num_obs[b] > t) ? 1 : 0;
            float denom = fmaxf((float)sm * (float)Dsz, 1.0f);
            nll += lp / denom;
            l1  += lp;
            l2  += kp;
        }
        nll /= (float)Tsz;
        out[Bsz * Hsz + 0] = l1 + 0.1f * l2;
        out[Bsz * Hsz + 1] = nll;
        out[Bsz * Hsz + 2] = l1;
        out[Bsz * Hsz + 3] = l2;
    }
}

// ---------------------------------------------------------------- launcher
extern "C" void kernel_launch(void* const* d_in, const int* in_sizes, int n_in,
                              void* d_out, int out_size, void* d_ws, size_t ws_size,
                              hipStream_t stream) {
    (void)in_sizes; (void)n_in; (void)out_size; (void)ws_size;
    const float* times   = (const float*)d_in[0];
    const float* Xg      = (const float*)d_in[1];
    const float* Mg      = (const float*)d_in[2];
    const float* cov     = (const float*)d_in[3];
    // d_in[4] = delta_t (unused by the reference math)
    const float* cov_w1  = (const float*)d_in[5];
    const float* cov_b1  = (const float*)d_in[6];
    const float* cov_w2  = (const float*)d_in[7];
    const float* cov_b2  = (const float*)d_in[8];
    const float* p_w1    = (const float*)d_in[9];
    const float* p_b1    = (const float*)d_in[10];
    const float* p_w2    = (const float*)d_in[11];
    const float* p_b2    = (const float*)d_in[12];
    const float* ode_whr = (const float*)d_in[13];
    const float* ode_whz = (const float*)d_in[14];
    const float* ode_whh = (const float*)d_in[15];
    const float* wprep   = (const float*)d_in[16];
    const float* bprep   = (const float*)d_in[17];
    const float* gru_wih = (const float*)d_in[18];
    const float* gru_whh = (const float*)d_in[19];
    const float* gru_bih = (const float*)d_in[20];
    const float* gru_bhh = (const float*)d_in[21];
    const int*   num_obs = (const int*)  d_in[22];

    float* ws = (float*)d_ws;
    float* h     = ws;                 // 512*512                = 262144 floats
    v2f* whrP    = (v2f*)(ws + 262144);  // 512x512  -> 131072 v2f (262144 f)
    v2f* whzP    = (v2f*)(ws + 524288);
    v2f* whhP    = (v2f*)(ws + 786432);
    v2f* whhGP   = (v2f*)(ws + 1048576); // 512x1536 -> 393216 v2f (786432 f)
    v2f* wihP    = (v2f*)(ws + 1835008); // 1024x1536-> 786432 v2f (1572864 f)
    v2f* pw1P    = (v2f*)(ws + 3407872); // 512x128  -> 32768 v2f  (65536 f)
    v2f* pw2P    = (v2f*)(ws + 3473408); // 128x64   -> 4096 v2f   (8192 f)
    v2f* cw1P    = (v2f*)(ws + 3481600); // 64x128   -> 4096 v2f   (8192 f)
    v2f* cw2P    = (v2f*)(ws + 3489792); // 128x512  -> 32768 v2f  (65536 f)
    float* lastt    = ws + 3555328;      // 512
    float* loss_par = ws + 3555840;      // 64*32
    float* kl_par   = ws + 3557888;      // 64*32

    k_zero<<<1, 512, 0, stream>>>(lastt);
    k_packT<<< 512, 256, 0, stream>>>(whrP,  ode_whr, 512,  512);
    k_packT<<< 512, 256, 0, stream>>>(whzP,  ode_whz, 512,  512);
    k_packT<<< 512, 256, 0, stream>>>(whhP,  ode_whh, 512,  512);
    k_packT<<<1536, 256, 0, stream>>>(whhGP, gru_whh, 1536, 512);
    k_packT<<<3072, 256, 0, stream>>>(wihP,  gru_wih, 1536, 1024);
    k_packB<<< 128, 256, 0, stream>>>(pw1P, p_w1,   512, 128);
    k_packB<<<  16, 256, 0, stream>>>(pw2P, p_w2,   128, 64);
    k_packB<<<  16, 256, 0, stream>>>(cw1P, cov_w1,  64, 128);
    k_packB<<< 128, 256, 0, stream>>>(cw2P, cov_w2, 128, 512);

    k_h0<<<32, 512, 0, stream>>>(cov, cw1P, cov_b1, cw2P, cov_b2, h);

    for (int t = 0; t < Tsz; ++t) {
        k_step<<<32, 512, 0, stream>>>(t, times, Xg, Mg, num_obs,
                                       pw1P, p_b1, pw2P, p_b2,
                                       wprep, bprep, gru_bih, gru_bhh,
                                       whrP, whzP, whhP, whhGP, wihP,
                                       h, lastt, loss_par, kl_par);
    }
    k_final<<<256, 256, 0, stream>>>(h, loss_par, kl_par, num_obs, (float*)d_out);
}